// LoopyBeliefPropagation_46883863003180
// MI455X (gfx1250) — compile-verified
//
#include <hip/hip_runtime.h>
#include <stdint.h>

#define LBP_EPS 1e-8f
#define LBP_ITERS 10
#define BLK 256

// ---------------------------------------------------------------------------
// gfx1250 async global->LDS helpers (ASYNCcnt path). Guarded so the host pass
// and non-gfx1250 arches use a plain load fallback.
// ---------------------------------------------------------------------------
#if defined(__gfx1250__)
typedef int lbp_v4i __attribute__((vector_size(4 * sizeof(int))));
typedef __attribute__((address_space(1))) lbp_v4i* lbp_gptr_v4i;
typedef __attribute__((address_space(3))) lbp_v4i* lbp_lptr_v4i;

__device__ __forceinline__ void lbp_async_copy_b128(const void* g, void* l) {
#if __has_builtin(__builtin_amdgcn_global_load_async_to_lds_b128)
  __builtin_amdgcn_global_load_async_to_lds_b128(
      (lbp_gptr_v4i)(g), (lbp_lptr_v4i)(l), 0, 0);
#else
  unsigned lds_off = (unsigned)(uintptr_t)(l);  // low 32 bits of generic LDS ptr = LDS offset
  asm volatile("global_load_async_to_lds_b128 %0, %1, off"
               :: "v"(lds_off), "v"(g) : "memory");
#endif
}
__device__ __forceinline__ void lbp_wait_async0() {
#if __has_builtin(__builtin_amdgcn_s_wait_asynccnt)
  __builtin_amdgcn_s_wait_asynccnt(0);
  asm volatile("" ::: "memory");
#else
  asm volatile("s_wait_asynccnt 0" ::: "memory");
#endif
}
#endif

// ---------------------------------------------------------------------------
// Kernel B (placed first so the disasm snippet shows the async path):
//   prod = exp(logphi[src] + node_log[src] - logmsg[rev])
//   m[j] = sum_i psi[e,i,j] * prod[i];  msg[e] = m / (sum(m)+eps)
// psi (contiguous 16B/edge) is staged via async global->LDS copy; the
// s_wait_asynccnt sits inside the guarded block AFTER the gathers/exp so the
// async fetch overlaps the random src/rev gathers.
// ---------------------------------------------------------------------------
__global__ void lbp_edge_update(const float* __restrict__ logphi,
                                const float* __restrict__ psi,
                                const int* __restrict__ srcv,
                                const int* __restrict__ rev,
                                const float* __restrict__ node_log,
                                const float* __restrict__ logmsg,
                                float* __restrict__ msg,
                                int E) {
  __shared__ float4 spsi[BLK];
  int e  = blockIdx.x * BLK + threadIdx.x;
  int ee = (e < E) ? e : (E - 1);

#if defined(__gfx1250__)
  lbp_async_copy_b128(psi + 4ull * (unsigned long long)ee, &spsi[threadIdx.x]);
#else
  spsi[threadIdx.x] = ((const float4*)psi)[ee];
#endif

  if (e < E) {
    int s = srcv[e];
    int r = rev[e];
    float2 lp = ((const float2*)logphi)[s];
    float2 nl = ((const float2*)node_log)[s];
    float2 lm = ((const float2*)logmsg)[r];
    float p0 = __expf(lp.x + nl.x - lm.x);
    float p1 = __expf(lp.y + nl.y - lm.y);

#if defined(__gfx1250__)
    lbp_wait_async0();
#endif
    float4 q = spsi[threadIdx.x];  // {psi00, psi01, psi10, psi11}

    float m0 = fmaf(q.x, p0, q.z * p1);
    float m1 = fmaf(q.y, p0, q.w * p1);
    float inv = __builtin_amdgcn_rcpf(m0 + m1 + LBP_EPS);
    ((float2*)msg)[e] = make_float2(m0 * inv, m1 * inv);
  }
}

// ---------------------------------------------------------------------------
// Kernel A: logmsg[e] = log(msg[e]);  node_log[dst[e]] += logmsg[e]  (atomic)
// ---------------------------------------------------------------------------
__global__ void lbp_log_scatter(const float* __restrict__ msg,
                                const int* __restrict__ dstv,
                                float* __restrict__ logmsg,
                                float* __restrict__ node_log,
                                int E) {
  int e = blockIdx.x * BLK + threadIdx.x;
  if (e >= E) return;
  float2 mv = ((const float2*)msg)[e];
  float l0 = __logf(mv.x);
  float l1 = __logf(mv.y);
  ((float2*)logmsg)[e] = make_float2(l0, l1);
  int d = dstv[e];
  atomicAdd(&node_log[2 * d + 0], l0);
  atomicAdd(&node_log[2 * d + 1], l1);
}

// ---------------------------------------------------------------------------
// small utility kernels
// ---------------------------------------------------------------------------
__global__ void lbp_fill_f32(float* __restrict__ p, float v, int n) {
  int i = blockIdx.x * BLK + threadIdx.x;
  if (i < n) p[i] = v;
}

__global__ void lbp_log_copy(const float* __restrict__ in, float* __restrict__ out, int n) {
  int i = blockIdx.x * BLK + threadIdx.x;
  if (i < n) out[i] = __logf(in[i]);
}

// ---------------------------------------------------------------------------
// Final beliefs: b = phi * exp(node_log); normalize
// ---------------------------------------------------------------------------
__global__ void lbp_beliefs(const float* __restrict__ phi,
                            const float* __restrict__ node_log,
                            float* __restrict__ out,
                            int N) {
  int i = blockIdx.x * BLK + threadIdx.x;
  if (i >= N) return;
  float2 ph = ((const float2*)phi)[i];
  float2 nl = ((const float2*)node_log)[i];
  float b0 = ph.x * __expf(nl.x);
  float b1 = ph.y * __expf(nl.y);
  float inv = __builtin_amdgcn_rcpf(b0 + b1 + LBP_EPS);
  ((float2*)out)[i] = make_float2(b0 * inv, b1 * inv);
}

// ---------------------------------------------------------------------------
// launch
// ---------------------------------------------------------------------------
extern "C" void kernel_launch(void* const* d_in, const int* in_sizes, int n_in,
                              void* d_out, int out_size, void* d_ws, size_t ws_size,
                              hipStream_t stream) {
  const float* phi  = (const float*)d_in[0];   // [N,2]
  const float* psi  = (const float*)d_in[1];   // [E,2,2]
  const int*   edge = (const int*)d_in[2];     // [2,E]
  const int*   rev  = (const int*)d_in[3];     // [E]

  const int N = in_sizes[0] / 2;
  const int E = in_sizes[3];
  const int* srcv = edge;
  const int* dstv = edge + E;

  float* beliefs_out = (float*)d_out;                       // N*2 floats
  float* msg         = (float*)d_out + (size_t)2 * N;       // E*2 floats (output + working)

  float* logphi   = (float*)d_ws;                           // N*2
  float* node_log = logphi + (size_t)2 * N;                 // N*2
  float* logmsg   = node_log + (size_t)2 * N;               // E*2

  dim3 blk(BLK);
  dim3 gE((E + BLK - 1) / BLK);
  dim3 gE2((2 * E + BLK - 1) / BLK);
  dim3 gN((N + BLK - 1) / BLK);
  dim3 gN2((2 * N + BLK - 1) / BLK);

  // setup: logphi = log(phi), msg = 1
  lbp_log_copy<<<gN2, blk, 0, stream>>>(phi, logphi, 2 * N);
  lbp_fill_f32<<<gE2, blk, 0, stream>>>(msg, 1.0f, 2 * E);

  for (int it = 0; it < LBP_ITERS; ++it) {
    lbp_fill_f32<<<gN2, blk, 0, stream>>>(node_log, 0.0f, 2 * N);
    lbp_log_scatter<<<gE, blk, 0, stream>>>(msg, dstv, logmsg, node_log, E);
    lbp_edge_update<<<gE, blk, 0, stream>>>(logphi, psi, srcv, rev,
                                            node_log, logmsg, msg, E);
  }

  // final node_log from converged msg, then beliefs
  lbp_fill_f32<<<gN2, blk, 0, stream>>>(node_log, 0.0f, 2 * N);
  lbp_log_scatter<<<gE, blk, 0, stream>>>(msg, dstv, logmsg, node_log, E);
  lbp_beliefs<<<gN, blk, 0, stream>>>(phi, node_log, beliefs_out, N);
}